// MPLayer_56513179680842
// MI455X (gfx1250) — compile-verified
//
#include <hip/hip_runtime.h>

// SNN IF-neuron presim/correct/sim kernel for MI455X (gfx1250).
// Memory-bound streaming kernel: 16 MB in, 128 MB out -> ~6.2 us at 23.3 TB/s.
// Strategy: b128 coalesced NT loads/stores (CDNA5 TH=NT cache hints),
// gfx1250 global_prefetch_b8 for the next grid-stride chunk, VALU minimized
// (fmax-folded selects, SALU lane-mask spike tracking, FMA soft reset).

typedef float __attribute__((ext_vector_type(4))) v4f;

#define PRESIM 4
#define SIM    8

__global__ __launch_bounds__(256) void snn_if_stream_kernel(
    const float* __restrict__ x, float* __restrict__ out, int n4)
{
    const int stride = gridDim.x * blockDim.x;
    for (int g = blockIdx.x * blockDim.x + threadIdx.x; g < n4; g += stride) {
        // Prefetch next chunk for this thread (gfx1250 global_prefetch_b8).
        if (g + stride < n4) {
            __builtin_prefetch((const void*)(x + 4 * (size_t)(g + stride)), 0, 3);
        }

        // 16B non-temporal load: input is read exactly once.
        v4f xin = __builtin_nontemporal_load((const v4f*)x + g);

        v4f outv[SIM];

        #pragma unroll
        for (int j = 0; j < 4; ++j) {
            const float inp = xin[j] * 0.5f;   // x / v_threshold (V_TH = 2)

            // ---- presim phase: IF soft-reset dynamics + membrane stats
            float v    = 0.5f;
            float mmin = 1000.0f;
            float mmax = -1000.0f;
            bool  any_spike = false;           // lane-mask in SGPR: s_or_b32
            #pragma unroll
            for (int t = 0; t < PRESIM; ++t) {
                v += inp;
                const bool sp = (v >= 1.0f);
                // mem_seq min: where spiked, candidate == pre-reset v
                // (reference's v_post + 1 == v_pre when sp).
                const float mn = fminf(mmin, v);
                mmin = sp ? mn : mmin;
                // soft reset
                const float vm1 = v - 1.0f;
                v = sp ? vm1 : v;
                // mem_seq2 max: where not spiked, candidate == post-reset v
                const float mx = fmaxf(mmax, v);
                mmax = sp ? mmax : mx;
                any_spike = any_spike || sp;
            }

            // ---- membrane-potential correction (COS, opt_epoch == 0 branch)
            // where(d > 1, d, 1) == max_num(d, 1)  (NaN-free inputs)
            const float val  = fmaxf(mmin - 0.999f, 1.0f);
            const float val2 = fmaxf(1.001f - mmax, 1.0f);
            const bool  low  = (v < 0.0f) && any_spike;   // rate>0.001 <=> any spike
            float mem_init = low ? (0.5f - val) : 0.5f;
            if (v >= 1.0f) mem_init = 0.5f + val2;

            // ---- sim phase: 8 IF steps, spike * v_threshold
            float v2 = mem_init;
            #pragma unroll
            for (int t = 0; t < SIM; ++t) {
                v2 += inp;
                const bool sp = (v2 >= 1.0f);
                const float o = sp ? 2.0f : 0.0f;
                outv[t][j] = o;
                v2 = fmaf(o, -0.5f, v2);       // exact: o in {0,2} -> subtract spike
            }
        }

        // 8 x 16B non-temporal stores: output written once, never re-read here.
        // out[t*N + i] layout; per-wave each store is fully coalesced.
        #pragma unroll
        for (int t = 0; t < SIM; ++t) {
            __builtin_nontemporal_store(outv[t],
                                        (v4f*)out + (size_t)t * (size_t)n4 + g);
        }
    }
}

extern "C" void kernel_launch(void* const* d_in, const int* in_sizes, int n_in,
                              void* d_out, int out_size, void* d_ws, size_t ws_size,
                              hipStream_t stream) {
    const float* x = (const float*)d_in[0];
    float* out = (float*)d_out;

    const int n  = in_sizes[0];   // 32*128*32*32 = 4,194,304 (divisible by 4)
    const int n4 = n >> 2;        // float4 groups = 1,048,576

    // 1024 blocks x 256 threads (8 wave32/block): 4 grid-stride iterations per
    // thread so the next-chunk prefetch overlaps the 8 NT stores per iteration.
    const int threads = 256;
    const int blocks  = 1024;
    snn_if_stream_kernel<<<blocks, threads, 0, stream>>>(x, out, n4);
}